// EKFGradient_30597347017200
// MI455X (gfx1250) — compile-verified
//
#include <hip/hip_runtime.h>

// ---------------------------------------------------------------------------
// EKF loss for MI455X (gfx1250, wave32).
// The 6-state KF decouples into three independent 2-state filters; covariance
// (gains) are data-independent -> precompute table once, stream the 100MB of
// measurements once (bandwidth bound, ~4.3us floor at 23.3 TB/s) with async
// global->LDS copies, one (trajectory,pair) scalar recursion per lane.
// ---------------------------------------------------------------------------

#define EKF_DT (1.0f/120.0f)
#define TS 64                 // time steps per LDS tile
#define TRAJ_PER_BLK 32
#define ZROW_F 196            // padded LDS row: 64*3 + 4 floats (16B pad -> only 2-way bank conflict)
#define ZROW_B (ZROW_F*4)

typedef __attribute__((ext_vector_type(2))) float v2f;
typedef __attribute__((ext_vector_type(8))) float v8f;
typedef int v4i __attribute__((ext_vector_type(4)));
typedef __attribute__((address_space(1))) v4i glb_v4i;
typedef __attribute__((address_space(3))) v4i lds_v4i;
typedef __attribute__((address_space(3))) char lds_char;

// ---- CDNA5 async global->LDS copy (tracked by ASYNCcnt) --------------------
__device__ __forceinline__ void async_cp16(const void* g, void* l) {
#if __has_builtin(__builtin_amdgcn_global_load_async_to_lds_b128)
    __builtin_amdgcn_global_load_async_to_lds_b128((glb_v4i*)g, (lds_v4i*)l, 0, 0);
#else
    unsigned loff = (unsigned)(__SIZE_TYPE__)(lds_char*)l;
    asm volatile("global_load_async_to_lds_b128 %0, %1, off" :: "v"(loff), "v"(g) : "memory");
#endif
}

#if __has_builtin(__builtin_amdgcn_s_wait_asynccnt)
#define WAIT_ASYNC() __builtin_amdgcn_s_wait_asynccnt(0)
#else
#define WAIT_ASYNC() asm volatile("s_wait_asynccnt 0x0" ::: "memory")
#endif

// ---------------------------------------------------------------------------
// Kernel A: per-pair 2x2 Riccati recursion (data-independent).
// Lane p in {0,1,2} runs pair p; stores {k0, k1, 1/S, S} per (t,pair).
// Then the whole wave computes mean_t det(S_t) = mean_t Sx*Sy*Sth.
// ---------------------------------------------------------------------------
__global__ __launch_bounds__(32) void ekf_gains(const float* __restrict__ dyna,
                                                const float* __restrict__ cov,
                                                float* __restrict__ tab,
                                                float* __restrict__ aux,
                                                int T) {
    const int lane = threadIdx.x;
    const float d = EKF_DT;
    const float a = 1.0f - dyna[1] * d;
    if (lane < 3) {
        const float r  = cov[lane];
        const float qp = (lane == 2) ? cov[5] : cov[3];
        const float qv = (lane == 2) ? cov[6] : cov[4];
        float p00 = 0.01f, p01 = 0.0f, p11 = 0.01f;
        for (int t = 0; t < T; ++t) {
            float t1   = fmaf(d, p11, p01);              // p01 + d*p11
            float pp01 = a * t1;
            float pp00 = p00 + d * (p01 + t1) + qp;      // p00 + 2d*p01 + d^2*p11 + qp
            float pp11 = fmaf(a * a, p11, qv);
            float S  = pp00 + r;
            float rS = 1.0f / S;
            float k0 = pp00 * rS;
            float k1 = pp01 * rS;
            float rrS = r * rS;
            p00 = pp00 * rrS;
            p01 = pp01 * rrS;
            p11 = fmaf(-k1, pp01, pp11);
            float4 e; e.x = k0; e.y = k1; e.z = rS; e.w = S;
            ((float4*)tab)[t * 3 + lane] = e;
        }
    }
    asm volatile("s_wait_storecnt 0x0" ::: "memory");
    __syncthreads();
    // parallel sum of det(S_t) over the wave
    float dsum = 0.0f;
    for (int t = lane; t < T; t += 32) {
        float sx = tab[(t * 3 + 0) * 4 + 3];
        float sy = tab[(t * 3 + 1) * 4 + 3];
        float st = tab[(t * 3 + 2) * 4 + 3];
        dsum += sx * sy * st;
    }
    for (int m = 16; m >= 1; m >>= 1) dsum += __shfl_xor(dsum, m, 32);
    if (lane == 0) aux[0] = dsum / (float)T;             // mean_t det(S_t)
}

// ---------------------------------------------------------------------------
// Kernel B: main streaming pass. Block = 96 threads (3 waves), wave = pair,
// lane = trajectory. Double-buffered async LDS tiles of TS steps.
// ---------------------------------------------------------------------------
__device__ __forceinline__ void issue_tile(const float* data, const float* tab,
                                           int tid, int b0, size_t T, int t0,
                                           float* zb, float4* tb) {
    const int rowSel = tid / 48;            // 96 threads * 16B = 2 trajectory rows / iter
    const int within = (tid % 48) * 16;     // byte offset inside the 768B row
    const char* dbytes = (const char*)data;
    #pragma unroll
    for (int i = 0; i < 16; ++i) {
        int tl = 2 * i + rowSel;
        const char* g = dbytes + ((size_t)(b0 + tl) * T + (size_t)t0) * 12 + within;
        char* l = (char*)zb + (size_t)tl * ZROW_B + within;
        async_cp16(g, l);
    }
    const char* tbytes = (const char*)tab + (size_t)t0 * 48;   // (t0*3) float4 entries
    #pragma unroll
    for (int j = 0; j < 2; ++j) {
        int off = (j * 96 + tid) * 16;                         // TS*3*16 = 3072B total
        async_cp16(tbytes + off, (char*)tb + off);
    }
}

__global__ __launch_bounds__(96) void ekf_main(const float* __restrict__ data,
                                               const float* __restrict__ x0,
                                               const float* __restrict__ tab,
                                               const float* __restrict__ dyna,
                                               float* __restrict__ partials,
                                               int T) {
    __shared__ __align__(16) float zbuf[2][TRAJ_PER_BLK * ZROW_F];
    __shared__ float4 tbuf[2][TS * 3];
    __shared__ float red[4];

    const int tid  = threadIdx.x;
    const int pair = tid >> 5;          // wave index = pair (wave-uniform)
    const int l    = tid & 31;          // lane = trajectory within block
    const int b0   = blockIdx.x * TRAJ_PER_BLK;

    const float d = EKF_DT;
    const float a = 1.0f - dyna[1] * d;

    const int posIdx = (pair < 2) ? pair     : 4;
    const int velIdx = (pair < 2) ? pair + 2 : 5;
    float x = x0[(size_t)(b0 + l) * 6 + posIdx];
    float v = x0[(size_t)(b0 + l) * 6 + velIdx];
    float acc = 0.0f;

    const int nt = T / TS;
    issue_tile(data, tab, tid, b0, (size_t)T, 0, zbuf[0], tbuf[0]);
    WAIT_ASYNC();
    __syncthreads();

    for (int c = 0; c < nt; ++c) {
        const int cur = c & 1;
        if (c + 1 < nt)
            issue_tile(data, tab, tid, b0, (size_t)T, (c + 1) * TS, zbuf[cur ^ 1], tbuf[cur ^ 1]);

        const float*  zr = &zbuf[cur][l * ZROW_F + pair];
        const float4* tr = tbuf[cur] + pair;    // uniform per wave -> ds broadcast
        #pragma unroll 8
        for (int tl = 0; tl < TS; ++tl) {
            float  z  = zr[tl * 3];
            float4 g  = tr[tl * 3];
            float  xp = fmaf(d, v, x);
            float  y  = z - xp;
            x   = fmaf(g.x, y, xp);
            v   = fmaf(g.y, y, a * v);
            acc = fmaf(y * g.z, y, acc);        // y^2 / S
        }
        if (c + 1 < nt) WAIT_ASYNC();
        __syncthreads();
    }

    for (int m = 16; m >= 1; m >>= 1) acc += __shfl_xor(acc, m, 32);
    if (l == 0) red[pair] = acc;
    __syncthreads();
    if (tid == 0) partials[blockIdx.x] = red[0] + red[1] + red[2];
}

// ---------------------------------------------------------------------------
// Kernel C: deterministic final reduction of the per-block partials using the
// matrix pipe: D = ones(16x4) x B(4x16) + C gives column sums per WMMA.
// ---------------------------------------------------------------------------
__global__ __launch_bounds__(32) void ekf_final(const float* __restrict__ partials,
                                                const float* __restrict__ aux,
                                                float* __restrict__ out,
                                                int nb, int Btr, int T) {
    const int lane = threadIdx.x;
    v2f A; A.x = 1.0f; A.y = 1.0f;              // ones 16x4 A-matrix
    v8f C = {};
    for (int k = 0; k < nb; k += 64) {          // 64 partials per WMMA (B is 4x16)
        v2f Bv; Bv.x = partials[k + lane]; Bv.y = partials[k + 32 + lane];
        C = __builtin_amdgcn_wmma_f32_16x16x4_f32(false, A, false, Bv,
                                                  (short)0, C, false, false);
    }
    float colsum = C[0];                        // every row of D equals the column sums
    colsum += __shfl_xor(colsum, 1, 32);
    colsum += __shfl_xor(colsum, 2, 32);
    colsum += __shfl_xor(colsum, 4, 32);
    colsum += __shfl_xor(colsum, 8, 32);        // lanes 0..15: sum over 16 columns
    if (lane == 0)
        out[0] = aux[0] + colsum / ((float)Btr * (float)T);
}

// ---------------------------------------------------------------------------
extern "C" void kernel_launch(void* const* d_in, const int* in_sizes, int n_in,
                              void* d_out, int out_size, void* d_ws, size_t ws_size,
                              hipStream_t stream) {
    const float* data = (const float*)d_in[0];   // [B,T,3]
    const float* init = (const float*)d_in[1];   // [B,6]
    const float* dyna = (const float*)d_in[2];   // [3]
    const float* cov  = (const float*)d_in[3];   // [7]

    const int Btr = in_sizes[1] / 6;             // 4096
    const int T   = in_sizes[0] / (3 * Btr);     // 2048 (multiple of TS)

    float* tab      = (float*)d_ws;              // T*3 float4 gain entries (96KB)
    float* aux      = tab + (size_t)T * 12;      // [0] = mean_t det(S_t)
    float* partials = aux + 1;                   // nblk per-block partials

    const int nblk = Btr / TRAJ_PER_BLK;         // 128

    ekf_gains<<<1, 32, 0, stream>>>(dyna, cov, tab, aux, T);
    ekf_main<<<nblk, 96, 0, stream>>>(data, init, tab, dyna, partials, T);
    ekf_final<<<1, 32, 0, stream>>>(partials, aux, (float*)d_out, nblk, Btr, T);
}